// FWPCell_82411832476056
// MI455X (gfx1250) — compile-verified
//
#include <hip/hip_runtime.h>

typedef __attribute__((ext_vector_type(2))) float v2f;
typedef __attribute__((ext_vector_type(8))) float v8f;

#define NB 65536            // batch
#define WAVES_PER_BLOCK1 4
#define ROWS_PER_WAVE 16

// ---------------------------------------------------------------------------
// Kernel 1: res = X@W_enc^T+b_enc ; [qubit|layer] = res@[W_qubit^T|W_layer^T]+b
//           params = layer[d]*qubit[q] + prev   (written to d_out + NB)
// Uses V_WMMA_F32_16X16X4_F32 (exact f32, matches reference numerics).
// All padded-lane loads use clamped (always-valid) addresses so they lower to
// unconditional global_load + v_cndmask instead of divergent exec branches.
// ---------------------------------------------------------------------------
__global__ __launch_bounds__(128) void linear_params_kernel(
    const float* __restrict__ X,        // [B,8]
    const float* __restrict__ prev,     // [B,2,8]
    const float* __restrict__ W_enc,    // [8,8]
    const float* __restrict__ b_enc,    // [8]
    const float* __restrict__ W_layer,  // [2,8]
    const float* __restrict__ b_layer,  // [2]
    const float* __restrict__ W_qubit,  // [8,8]
    const float* __restrict__ b_qubit,  // [8]
    float* __restrict__ out)            // [NB] out | [NB,16] params
{
    __shared__ float sres[WAVES_PER_BLOCK1][16][9];   // pad 9: conflict-free cols
    const int tid = threadIdx.x;
    const int w = tid >> 5;
    const int l = tid & 31;
    const int h = l >> 4;      // lane half (selects K pair / M+8)
    const int n = l & 15;      // D/B column, A row
    const bool nv8  = (n < 8);
    const bool nv10 = (n < 10);
    const int row0 = (blockIdx.x * WAVES_PER_BLOCK1 + w) * ROWS_PER_WAVE;

    // ---- GEMM1: D = X(16x8) @ W_enc^T(8x16,padded) + b_enc  (2 k-steps) ----
    const float be = nv8 ? b_enc[n & 7] : 0.0f;       // clamped load + select
    v8f acc = {be, be, be, be, be, be, be, be};
    const float* __restrict__ we = W_enc + (n & 7) * 8;   // always-valid base
#pragma unroll
    for (int s = 0; s < 2; ++s) {
        const int k0 = 4 * s + 2 * h;                 // K for VGPR0 of A/B
        v2f a, b;
        const float* xr = X + (size_t)(row0 + n) * 8; // A row = n
        a.x = xr[k0];
        a.y = xr[k0 + 1];
        const float w0 = we[k0];                      // unconditional loads
        const float w1 = we[k0 + 1];
        b.x = nv8 ? w0 : 0.0f;                        // v_cndmask, no branch
        b.y = nv8 ? w1 : 0.0f;
        acc = __builtin_amdgcn_wmma_f32_16x16x4_f32(false, a, false, b,
                                                    (short)0, acc, false, false);
    }
    // D-layout -> row-major res tile in LDS (wave-private slab)
#pragma unroll
    for (int v = 0; v < 8; ++v)
        if (nv8) sres[w][v + 8 * h][n] = acc[v];
    __syncthreads();

    // ---- GEMM2: fused [qubit_idx (n<8) | layer_idx (n=8,9)] ----
    const float bq = b_qubit[n & 7];
    const float bl = b_layer[n & 1];
    const float b2 = nv8 ? bq : (nv10 ? bl : 0.0f);
    v8f acc2 = {b2, b2, b2, b2, b2, b2, b2, b2};
    // Fused B base pointer: W_qubit row n (n<8) else W_layer row n&1 (valid for all n)
    const float* __restrict__ w2 = nv8 ? (W_qubit + n * 8) : (W_layer + (n & 1) * 8);
#pragma unroll
    for (int s = 0; s < 2; ++s) {
        const int k0 = 4 * s + 2 * h;
        v2f a, b;
        a.x = sres[w][n][k0];
        a.y = sres[w][n][k0 + 1];
        const float w0 = w2[k0];                      // unconditional loads
        const float w1 = w2[k0 + 1];
        b.x = nv10 ? w0 : 0.0f;
        b.y = nv10 ? w1 : 0.0f;
        acc2 = __builtin_amdgcn_wmma_f32_16x16x4_f32(false, a, false, b,
                                                     (short)0, acc2, false, false);
    }

    // ---- params[m][d][q] = layer[d]*qubit[q] + prev ----
    float* params = out + NB;
#pragma unroll
    for (int v = 0; v < 8; ++v) {
        const float qv = acc2[v];
        const float ld0 = __shfl(qv, (l & 16) + 8, 32);  // layer_idx[m][0]
        const float ld1 = __shfl(qv, (l & 16) + 9, 32);  // layer_idx[m][1]
        if (nv8) {
            const int m = row0 + v + 8 * h;
            params[m * 16 + n]     = ld0 * qv + prev[m * 16 + n];
            params[m * 16 + 8 + n] = ld1 * qv + prev[m * 16 + 8 + n];
        }
    }
}

// ---------------------------------------------------------------------------
// Kernel 2: 256-amplitude statevector per wave32. Amplitude index i[7:0]:
// lane = i[7:3] (qubits 0..4), reg = i[2:0] (qubits 5..7).
// ---------------------------------------------------------------------------
template <int XM>   // RY on a lane bit: partner = lane ^ XM
__device__ __forceinline__ void ry_lane(float psi[8], float c, float s, int lane) {
    const float sgn = (lane & XM) ? s : -s;   // bit=0: c*a0-s*a1 ; bit=1: c*a1+s*a0
#pragma unroll
    for (int r = 0; r < 8; ++r) {
        const float p = __shfl_xor(psi[r], XM, 32);
        psi[r] = c * psi[r] + sgn * p;
    }
}

template <int RB>   // RY on a register bit
__device__ __forceinline__ void ry_reg(float psi[8], float c, float s) {
#pragma unroll
    for (int r = 0; r < 8; ++r)
        if ((r & RB) == 0) {
            const float a0 = psi[r], a1 = psi[r | RB];
            psi[r]      = c * a0 - s * a1;
            psi[r | RB] = s * a0 + c * a1;
        }
}

template <int CM, int TX>   // CNOT: control lane-bit CM, target lane-xor TX
__device__ __forceinline__ void cnot_ll(float psi[8], int lane) {
    const bool hi = (lane & CM) != 0;
#pragma unroll
    for (int r = 0; r < 8; ++r) {
        const float p = __shfl_xor(psi[r], TX, 32);
        psi[r] = hi ? p : psi[r];
    }
}

template <int CM, int RB>   // CNOT: control lane-bit CM, target reg-bit RB
__device__ __forceinline__ void cnot_lr(float psi[8], int lane) {
    const bool hi = (lane & CM) != 0;
#pragma unroll
    for (int r = 0; r < 8; ++r)
        if ((r & RB) == 0) {
            const float a = psi[r], b = psi[r | RB];
            psi[r]      = hi ? b : a;
            psi[r | RB] = hi ? a : b;
        }
}

__global__ __launch_bounds__(256) void simulate_kernel(
    const float* __restrict__ X,       // [B,8] encoding angles
    const float* __restrict__ W_post,  // [1,4]
    const float* __restrict__ b_post,  // [1]
    float* __restrict__ out)           // [NB] out ; params at out+NB (from k1)
{
    const int item = (blockIdx.x * blockDim.x + threadIdx.x) >> 5;
    const int lane = threadIdx.x & 31;
    const float* params = out + NB + (size_t)item * 16;

    // Distributed sincos: lane g holds (c,s) for gate g (24 gates total).
    // Both candidate loads are unconditional (clamped index) -> cndmask select.
    const float tx = X[(size_t)item * 8 + (lane & 7)];
    const float tp = params[lane & 15];
    const float theta = (lane < 8) ? tx : ((lane < 24) ? tp : 0.0f);
    float sv, cv;
    sincosf(0.5f * theta, &sv, &cv);

    float psi[8];
#pragma unroll
    for (int r = 0; r < 8; ++r) psi[r] = 0.0625f;   // H-layer on |0..0>: 2^-4

#define CG(g) __shfl(cv, (g), 32)
#define SG(g) __shfl(sv, (g), 32)

    // Encoding layer: qubit q -> flat bit 7-q (lane bit 4-q for q<=4, reg bit 7-q else)
    ry_lane<16>(psi, CG(0), SG(0), lane);
    ry_lane<8 >(psi, CG(1), SG(1), lane);
    ry_lane<4 >(psi, CG(2), SG(2), lane);
    ry_lane<2 >(psi, CG(3), SG(3), lane);
    ry_lane<1 >(psi, CG(4), SG(4), lane);
    ry_reg<4>(psi, CG(5), SG(5));
    ry_reg<2>(psi, CG(6), SG(6));
    ry_reg<1>(psi, CG(7), SG(7));

#pragma unroll
    for (int k = 0; k < 2; ++k) {
        const int g0 = 8 + 8 * k;
        cnot_ll<16, 8>(psi, lane);          // CNOT(0,1)
        cnot_ll<4, 2>(psi, lane);           // CNOT(2,3)
        cnot_lr<1, 4>(psi, lane);           // CNOT(4,5)
        { float t = psi[2]; psi[2] = psi[3]; psi[3] = t;      // CNOT(6,7):
          t = psi[6]; psi[6] = psi[7]; psi[7] = t; }          // reg renames
        cnot_ll<8, 4>(psi, lane);           // CNOT(1,2)
        cnot_ll<2, 1>(psi, lane);           // CNOT(3,4)
        { float t = psi[4]; psi[4] = psi[6]; psi[6] = t;      // CNOT(5,6)
          t = psi[5]; psi[5] = psi[7]; psi[7] = t; }
        ry_lane<16>(psi, CG(g0 + 0), SG(g0 + 0), lane);
        ry_lane<8 >(psi, CG(g0 + 1), SG(g0 + 1), lane);
        ry_lane<4 >(psi, CG(g0 + 2), SG(g0 + 2), lane);
        ry_lane<2 >(psi, CG(g0 + 3), SG(g0 + 3), lane);
        ry_lane<1 >(psi, CG(g0 + 4), SG(g0 + 4), lane);
        ry_reg<4>(psi, CG(g0 + 5), SG(g0 + 5));
        ry_reg<2>(psi, CG(g0 + 6), SG(g0 + 6));
        ry_reg<1>(psi, CG(g0 + 7), SG(g0 + 7));
    }
#undef CG
#undef SG

    // <Z_p>, p=0..3 -> lane bits 4,3,2,1 ; per-lane prob mass then signed butterflies
    float local = 0.0f;
#pragma unroll
    for (int r = 0; r < 8; ++r) local += psi[r] * psi[r];

    float z0 = (lane & 16) ? -local : local;
    float z1 = (lane & 8)  ? -local : local;
    float z2 = (lane & 4)  ? -local : local;
    float z3 = (lane & 2)  ? -local : local;
#pragma unroll
    for (int off = 16; off >= 1; off >>= 1) {
        z0 += __shfl_xor(z0, off, 32);
        z1 += __shfl_xor(z1, off, 32);
        z2 += __shfl_xor(z2, off, 32);
        z3 += __shfl_xor(z3, off, 32);
    }
    if (lane == 0)
        out[item] = z0 * W_post[0] + z1 * W_post[1] + z2 * W_post[2]
                  + z3 * W_post[3] + b_post[0];
}

// ---------------------------------------------------------------------------
extern "C" void kernel_launch(void* const* d_in, const int* in_sizes, int n_in,
                              void* d_out, int out_size, void* d_ws, size_t ws_size,
                              hipStream_t stream) {
    const float* X       = (const float*)d_in[0];
    const float* prev    = (const float*)d_in[1];
    const float* W_enc   = (const float*)d_in[2];
    const float* b_enc   = (const float*)d_in[3];
    const float* W_layer = (const float*)d_in[4];
    const float* b_layer = (const float*)d_in[5];
    const float* W_qubit = (const float*)d_in[6];
    const float* b_qubit = (const float*)d_in[7];
    const float* W_post  = (const float*)d_in[8];
    const float* b_post  = (const float*)d_in[9];
    float* out = (float*)d_out;

    // Kernel 1: 16 rows/wave, 4 waves/block -> 64 rows/block
    const int blocks1 = NB / (ROWS_PER_WAVE * WAVES_PER_BLOCK1);   // 1024
    linear_params_kernel<<<blocks1, 128, 0, stream>>>(
        X, prev, W_enc, b_enc, W_layer, b_layer, W_qubit, b_qubit, out);

    // Kernel 2: 1 item/wave, 8 items/block
    const int blocks2 = NB / 8;                                    // 8192
    simulate_kernel<<<blocks2, 256, 0, stream>>>(X, W_post, b_post, out);
}